// GraphProjection_35553739276665
// MI455X (gfx1250) — compile-verified
//
#include <hip/hip_runtime.h>
#include <stdint.h>

typedef __attribute__((ext_vector_type(2))) float v2f;
typedef __attribute__((ext_vector_type(8))) float v8f;
typedef __attribute__((ext_vector_type(4))) unsigned int v4u;
typedef __attribute__((ext_vector_type(8))) int v8i;
typedef __attribute__((ext_vector_type(4))) int v4i;

#define NBS   256   // reduction blocks
#define TPB   256   // threads per block (8 waves on wave32)
#define PPB   64    // points per block in the interp kernel
#define DFEAT 32    // feature channels == wave32 lanes

struct F3 { float x, y, z; };

// ---------------------------------------------------------------------------
// Kernel 1: partial column sums of the (N,3) point cloud over FULL 64-point
// chunks only (wave-uniform loop bound -> no exec-mask predication, fully
// unconditional coalesced loads). Sum-reduction on the matrix pipe:
// D = A x ones + C gives row sums of A, so one v_wmma_f32_16x16x4_f32 folds
// 64 floats into the accumulator. Element placement inside the A tile is
// irrelevant for a total sum. The <64-point tail is handled in kernel 2.
// ---------------------------------------------------------------------------
__global__ void k_sum_partial(const float* __restrict__ pts,
                              float* __restrict__ partial, int N) {
    const int lane = threadIdx.x & 31;
    const int wave = threadIdx.x >> 5;
    const int wavesPerBlock = blockDim.x >> 5;
    const int gwave  = blockIdx.x * wavesPerBlock + wave;
    const int nwaves = gridDim.x * wavesPerBlock;

    v8f ax = {}, ay = {}, az = {};
    const v2f ones = {1.0f, 1.0f};

    for (long long base = (long long)gwave * 64; base + 64 <= (long long)N;
         base += (long long)nwaves * 64) {
        const F3 a = *(const F3*)(pts + (base + lane) * 3);        // 12B/lane
        const F3 b = *(const F3*)(pts + (base + 32 + lane) * 3);   // 12B/lane
        const v2f AX = {a.x, b.x};
        const v2f AY = {a.y, b.y};
        const v2f AZ = {a.z, b.z};
        ax = __builtin_amdgcn_wmma_f32_16x16x4_f32(false, AX, false, ones,
                                                   (short)0, ax, false, false);
        ay = __builtin_amdgcn_wmma_f32_16x16x4_f32(false, AY, false, ones,
                                                   (short)0, ay, false, false);
        az = __builtin_amdgcn_wmma_f32_16x16x4_f32(false, AZ, false, ones,
                                                   (short)0, az, false, false);
    }

    // Per lane: sum the 8 C VGPRs -> lanes 0..15 hold sum(rows 0..7),
    // lanes 16..31 hold sum(rows 8..15), replicated across each half.
    float tx = 0.f, ty = 0.f, tz = 0.f;
#pragma unroll
    for (int r = 0; r < 8; ++r) { tx += ax[r]; ty += ay[r]; tz += az[r]; }
    tx = __shfl(tx, 0, 32) + __shfl(tx, 16, 32);
    ty = __shfl(ty, 0, 32) + __shfl(ty, 16, 32);
    tz = __shfl(tz, 0, 32) + __shfl(tz, 16, 32);

    __shared__ float red[3][TPB / 32];
    if (lane == 0) { red[0][wave] = tx; red[1][wave] = ty; red[2][wave] = tz; }
    __syncthreads();
    if (threadIdx.x == 0) {
        float sx = 0.f, sy = 0.f, sz = 0.f;
        for (int w = 0; w < wavesPerBlock; ++w) {
            sx += red[0][w]; sy += red[1][w]; sz += red[2][w];
        }
        partial[blockIdx.x * 3 + 0] = sx;
        partial[blockIdx.x * 3 + 1] = sy;
        partial[blockIdx.x * 3 + 2] = sz;
    }
}

// ---------------------------------------------------------------------------
// Kernel 2: reduce NBS partial sums (+ the <64-point tail) -> mean stats[0..2]
// ---------------------------------------------------------------------------
__global__ void k_finalize_mean(const float* __restrict__ partial,
                                const float* __restrict__ pts,
                                float* __restrict__ stats, int nb,
                                float invN, int tailStart, int N) {
    __shared__ float s[3][NBS];
    const int t = threadIdx.x;
    for (int c = 0; c < 3; ++c)
        s[c][t] = (t < nb) ? partial[t * 3 + c] : 0.0f;
    const int tp = tailStart + t;
    if (tp < N) {
        s[0][t] += pts[tp * 3 + 0];
        s[1][t] += pts[tp * 3 + 1];
        s[2][t] += pts[tp * 3 + 2];
    }
    __syncthreads();
    for (int off = NBS / 2; off > 0; off >>= 1) {
        if (t < off)
            for (int c = 0; c < 3; ++c) s[c][t] += s[c][t + off];
        __syncthreads();
    }
    if (t == 0) {
        stats[0] = s[0][0] * invN;
        stats[1] = s[1][0] * invN;
        stats[2] = s[2][0] * invN;
    }
}

// ---------------------------------------------------------------------------
// Kernel 3: partial max of squared norms of centered points
// ---------------------------------------------------------------------------
__global__ void k_max_partial(const float* __restrict__ pts,
                              const float* __restrict__ stats,
                              float* __restrict__ pmax, int N) {
    const float mx = stats[0], my = stats[1], mz = stats[2];
    float m = 0.0f;
    for (long long p = (long long)blockIdx.x * blockDim.x + threadIdx.x; p < N;
         p += (long long)gridDim.x * blockDim.x) {
        float dx = pts[p * 3 + 0] - mx;
        float dy = pts[p * 3 + 1] - my;
        float dz = pts[p * 3 + 2] - mz;
        m = fmaxf(m, fmaf(dx, dx, fmaf(dy, dy, dz * dz)));
    }
    __shared__ float s[TPB];
    s[threadIdx.x] = m;
    __syncthreads();
    for (int off = TPB / 2; off > 0; off >>= 1) {
        if (threadIdx.x < off)
            s[threadIdx.x] = fmaxf(s[threadIdx.x], s[threadIdx.x + off]);
        __syncthreads();
    }
    if (threadIdx.x == 0) pmax[blockIdx.x] = s[0];
}

// ---------------------------------------------------------------------------
// Kernel 4: reduce partial maxes -> scale, inv_scale (stats[3] = 1/scale)
// ---------------------------------------------------------------------------
__global__ void k_finalize_scale(const float* __restrict__ pmax,
                                 float* __restrict__ stats, int nb) {
    __shared__ float s[NBS];
    const int t = threadIdx.x;
    s[t] = (t < nb) ? pmax[t] : 0.0f;
    __syncthreads();
    for (int off = NBS / 2; off > 0; off >>= 1) {
        if (t < off) s[t] = fmaxf(s[t], s[t + off]);
        __syncthreads();
    }
    if (t == 0) {
        float sc = sqrtf(s[0]);
        stats[4] = sc;
        stats[3] = 1.0f / sc;
    }
}

// ---------------------------------------------------------------------------
// Kernel 5: trilinear gather. One wave = one point, lane = feature channel,
// so the 8 corner gathers are 8 fully coalesced 128B loads per wave.
// Block's PPB point coords staged into LDS with the Tensor Data Mover.
// ---------------------------------------------------------------------------
__global__ void k_interp(const float* __restrict__ pts,
                         const float* __restrict__ voxel,
                         const int* __restrict__ vres,
                         const float* __restrict__ stats,
                         float* __restrict__ out, int N) {
    __shared__ float sc[PPB * 3];
    const int lane = threadIdx.x & 31;
    const int wave = threadIdx.x >> 5;
    const long long blockStart = (long long)blockIdx.x * PPB;

    if (threadIdx.x == 0) {
        // ---- Tensor DMA descriptor (D#): 1-D tile of PPB*3 f32 -> LDS ----
        unsigned long long gaddr =
            (unsigned long long)(uintptr_t)(pts + blockStart * 3);
        unsigned int ldsoff = (unsigned int)(uintptr_t)(&sc[0]); // low 32b = LDS offset
        long long remain = (long long)N * 3 - blockStart * 3;
        unsigned int tdim0 =
            (remain >= (long long)(PPB * 3)) ? (unsigned)(PPB * 3)
                                             : (remain > 0 ? (unsigned)remain : 0u);
        v4u g0;
        g0.x = 1u;                                    // count=1, user descriptor
        g0.y = ldsoff;                                // lds_addr (bytes)
        g0.z = (unsigned)(gaddr & 0xFFFFFFFFull);     // global_addr[31:0]
        g0.w = (unsigned)((gaddr >> 32) & 0x1FFFFFFull) | 0x80000000u; // addr[56:32] | type=2
        v8i g1;
        g1[0] = 0x00020000;                                   // data_size=2 (4B), no multicast
        g1[1] = (int)((tdim0 & 0xFFFFu) << 16);               // tensor_dim0[15:0]
        g1[2] = (int)(((tdim0 >> 16) & 0xFFFFu) | (1u << 16));// tensor_dim0[31:16], tensor_dim1=1
        g1[3] = (int)((unsigned)(PPB * 3) << 16);             // tile_dim0 = PPB*3
        g1[4] = 1;                                            // tile_dim1=1, tile_dim2=0
        g1[5] = PPB * 3;                                      // tensor_dim0_stride lo
        g1[6] = 0;                                            // stride hi / dim1_stride lo
        g1[7] = 0;                                            // dim1_stride hi
        v4i g2 = {0, 0, 0, 0};
        v4i g3 = {0, 0, 0, 0};
        v8i g4 = {0, 0, 0, 0, 0, 0, 0, 0};
        __builtin_amdgcn_tensor_load_to_lds(g0, g1, g2, g3, g4, 0);
        __builtin_amdgcn_s_wait_tensorcnt(0);
    }
    __syncthreads();

    const int   R   = vres[0];
    const float mx  = stats[0], my = stats[1], mz = stats[2];
    const float inv = stats[3];
    const float half_span = 0.5f * (float)(R - 1);
    const int   sx = R * R * DFEAT;   // x stride in elements
    const int   sy = R * DFEAT;       // y stride

    const int wavesPerBlock = blockDim.x >> 5;
    const int ptsPerWave = PPB / wavesPerBlock;
    for (int i = 0; i < ptsPerWave; ++i) {
        const int lp = wave * ptsPerWave + i;
        const long long p = blockStart + lp;
        if (p >= N) break;

        const float x = sc[lp * 3 + 0];
        const float y = sc[lp * 3 + 1];
        const float z = sc[lp * 3 + 2];

        const float cx = ((x - mx) * inv + 1.0f) * half_span;
        const float cy = ((y - my) * inv + 1.0f) * half_span;
        const float cz = ((z - mz) * inv + 1.0f) * half_span;

        const float lx = floorf(cx), hx = ceilf(cx);
        const float ly = floorf(cy), hy = ceilf(cy);
        const float lz = floorf(cz), hz = ceilf(cz);
        const float wx = (cx - lx) / (hx - lx);
        const float wy = (cy - ly) / (hy - ly);
        const float wz = (cz - lz) / (hz - lz);

        int ix0 = min(max((int)lx, 0), R - 1), ix1 = min(max((int)hx, 0), R - 1);
        int iy0 = min(max((int)ly, 0), R - 1), iy1 = min(max((int)hy, 0), R - 1);
        int iz0 = min(max((int)lz, 0), R - 1), iz1 = min(max((int)hz, 0), R - 1);

        const int bx0 = ix0 * sx, bx1 = ix1 * sx;
        const int by0 = iy0 * sy, by1 = iy1 * sy;
        const int bz0 = iz0 * DFEAT + lane, bz1 = iz1 * DFEAT + lane;

        // 8 coalesced 128B gathers (issued back-to-back for MLP)
        const float q000 = voxel[bx0 + by0 + bz0];
        const float q001 = voxel[bx0 + by0 + bz1];
        const float q010 = voxel[bx0 + by1 + bz0];
        const float q011 = voxel[bx0 + by1 + bz1];
        const float q100 = voxel[bx1 + by0 + bz0];
        const float q101 = voxel[bx1 + by0 + bz1];
        const float q110 = voxel[bx1 + by1 + bz0];
        const float q111 = voxel[bx1 + by1 + bz1];

        // exact nested-lerp order of the reference
        const float q00 = q000 * (1.0f - wx) + q100 * wx;
        const float q01 = q001 * (1.0f - wx) + q101 * wx;
        const float q10 = q010 * (1.0f - wx) + q110 * wx;
        const float q11 = q011 * (1.0f - wx) + q111 * wx;
        const float q0  = q00 * (1.0f - wy) + q10 * wy;
        const float q1  = q01 * (1.0f - wy) + q11 * wy;
        const float o   = q0 * (1.0f - wz) + q1 * wz;

        float* row = out + p * 35;
        row[3 + lane] = o;
        if (lane < 3) row[lane] = sc[lp * 3 + lane];
    }
}

// ---------------------------------------------------------------------------
// Host launcher
// ws layout (floats): [0..2]=mean, [3]=1/scale, [4]=scale,
//                     [8 .. 8+3*NBS) = partial sums, then NBS partial maxes.
// ---------------------------------------------------------------------------
extern "C" void kernel_launch(void* const* d_in, const int* in_sizes, int n_in,
                              void* d_out, int out_size, void* d_ws, size_t ws_size,
                              hipStream_t stream) {
    const float* pts   = (const float*)d_in[0];
    const float* voxel = (const float*)d_in[1];
    const int*   vres  = (const int*)d_in[2];
    const int N = in_sizes[0] / 3;

    float* out   = (float*)d_out;
    float* ws    = (float*)d_ws;
    float* stats = ws;               // [0..4]
    float* psum  = ws + 8;           // 3*NBS
    float* pmax  = ws + 8 + 3 * NBS; // NBS

    const int tailStart = (N / 64) * 64;
    k_sum_partial<<<NBS, TPB, 0, stream>>>(pts, psum, N);
    k_finalize_mean<<<1, NBS, 0, stream>>>(psum, pts, stats, NBS,
                                           1.0f / (float)N, tailStart, N);
    k_max_partial<<<NBS, TPB, 0, stream>>>(pts, stats, pmax, N);
    k_finalize_scale<<<1, NBS, 0, stream>>>(pmax, stats, NBS);

    const int nblocks = (N + PPB - 1) / PPB;
    k_interp<<<nblocks, TPB, 0, stream>>>(pts, voxel, vres, stats, out, N);
}